// EmbeddingLayer_59837484368478
// MI455X (gfx1250) — compile-verified
//
#include <hip/hip_runtime.h>
#include <hip/hip_bf16.h>

// Embedding gather: out[tok, :] = table[ids[tok], :], EMB = 768 fp32.
//
// MI455X reasoning:
//  - Zero FLOPs -> pure bandwidth problem, floor ~4.3us at 23.3 TB/s.
//  - Table (50257*768*4B ~= 154MB) fits in the 192MB global L2, so gather
//    reads hit L2 once warm. Keep table loads temporal (default).
//  - Output (48MiB) is write-once streaming: store with the non-temporal
//    hint (global_store_b128 th:NT) so it does not evict the table from L2.
//  - wave32: one wave per token row; 768 floats = 192 x b128; 6 x float4
//    per lane, all coalesced, 6 loads in flight to cover latency.

typedef __attribute__((ext_vector_type(4))) float v4f;

#ifndef EMB_DIM
#define EMB_DIM 768
#endif

__global__ __launch_bounds__(256) void
embedding_gather_kernel(const int* __restrict__ ids,
                        const float* __restrict__ table,
                        float* __restrict__ out,
                        int ntok) {
    const int gtid = blockIdx.x * blockDim.x + threadIdx.x;
    const int tok  = gtid >> 5;            // one wave32 per token row
    const int lane = threadIdx.x & 31;
    if (tok >= ntok) return;

    // Wave-uniform index -> force into SGPR so table row base is scalar and
    // the loads take the saddr (SGPR-base + VGPR-offset) form.
    const int row = __builtin_amdgcn_readfirstlane(ids[tok]);

    const v4f* __restrict__ src =
        reinterpret_cast<const v4f*>(table + (size_t)row * EMB_DIM) + lane;
    v4f* __restrict__ dst =
        reinterpret_cast<v4f*>(out + (size_t)tok * EMB_DIM) + lane;

    // 192 float4 per row / 32 lanes = 6 per lane. Load all six first (6
    // global_load_b128 in flight), then stream out with NT stores.
    v4f r0, r1, r2, r3, r4, r5;
    r0 = src[0 * 32];
    r1 = src[1 * 32];
    r2 = src[2 * 32];
    r3 = src[3 * 32];
    r4 = src[4 * 32];
    r5 = src[5 * 32];

    __builtin_nontemporal_store(r0, dst + 0 * 32);
    __builtin_nontemporal_store(r1, dst + 1 * 32);
    __builtin_nontemporal_store(r2, dst + 2 * 32);
    __builtin_nontemporal_store(r3, dst + 3 * 32);
    __builtin_nontemporal_store(r4, dst + 4 * 32);
    __builtin_nontemporal_store(r5, dst + 5 * 32);
}

extern "C" void kernel_launch(void* const* d_in, const int* in_sizes, int n_in,
                              void* d_out, int out_size, void* d_ws, size_t ws_size,
                              hipStream_t stream) {
    (void)n_in; (void)d_ws; (void)ws_size; (void)out_size;

    const int*   ids   = (const int*)d_in[0];    // [B*S] token ids
    const float* table = (const float*)d_in[1];  // [VOCAB, EMB] fp32
    float*       out   = (float*)d_out;          // [B*S, EMB] fp32

    const int ntok = in_sizes[0];                // B*S = 16384

    // 8 waves (256 threads) per block, one wave per token row.
    const int waves_per_block = 256 / 32;
    const int grid = (ntok + waves_per_block - 1) / waves_per_block;

    embedding_gather_kernel<<<grid, 256, 0, stream>>>(ids, table, out, ntok);
}